// PhotonicInterferenceLayer_90804198572667
// MI455X (gfx1250) — compile-verified
//
#include <hip/hip_runtime.h>
#include <math.h>

// ---------------- problem constants ----------------
#define DIMC    16          // embed dim (complex)
#define FW      32          // interleaved real/imag feature width
#define FW_P    36          // padded LDS row (bank spread, 16B aligned)
#define NHEADS  4
#define RELMAX  64
#define P_LEN   524288
#define NBLK    256
#define TPB     256
#define NWAVES  (TPB/32)    // 8 waves (wave32)
#define CHUNK   32          // positions projected per wave per iteration
#define POS_PER_WAVE (P_LEN / (NBLK * NWAVES))   // 256
#define NITER   (POS_PER_WAVE / CHUNK)           // 8

// workspace layout (floats)
#define WS_Z    0           // current z, 32 floats (interleaved re/im)
#define WS_Q    32          // current q, 32 floats
#define WS_RTAB 64          // rel-bias table 129*4 floats (includes 0.5 scale)
#define WS_PART 1024        // block partials: [NBLK][NHEADS][10] (m,l,acc[8])

typedef float v2f __attribute__((ext_vector_type(2)));
typedef float v8f __attribute__((ext_vector_type(8)));

__device__ __forceinline__ float mishf(float x) {
  return x * tanhf(log1pf(__expf(x)));
}

// ------------------------------------------------------------------
// init: z = candidate; q = Wq*z + bq; rtab[r][h] = 0.5 * <q_h, rel[r]_h>
// ------------------------------------------------------------------
__global__ __launch_bounds__(256) void pil_init(
    const float* __restrict__ cand, const float* __restrict__ Wq,
    const float* __restrict__ bq,   const float* __restrict__ rel,
    float* __restrict__ ws) {
  __shared__ float sz[FW];
  __shared__ float sq[FW];
  const int t = threadIdx.x;
  if (t < FW) sz[t] = cand[t];
  __syncthreads();
  if (t < FW) {
    float a = bq[t];
#pragma unroll
    for (int i = 0; i < FW; ++i) a += Wq[t * FW + i] * sz[i];
    sq[t] = a;
    ws[WS_Z + t] = sz[t];
    ws[WS_Q + t] = a;
  }
  __syncthreads();
  for (int e = t; e < (2 * RELMAX + 1) * NHEADS; e += 256) {
    const int r = e >> 2, h = e & 3;
    float a = 0.f;
#pragma unroll
    for (int u = 0; u < 8; ++u) a += sq[8 * h + u] * rel[r * FW + 8 * h + u];
    ws[WS_RTAB + e] = 0.5f * a;
  }
}

// ------------------------------------------------------------------
// scan: fused (Wk,Wv cache projection via f32 WMMA) + online softmax
// ------------------------------------------------------------------
__global__ __launch_bounds__(TPB) void pil_scan(
    const float* __restrict__ zp,
    const float* __restrict__ Wk, const float* __restrict__ bk,
    const float* __restrict__ Wv, const float* __restrict__ bv,
    float* __restrict__ ws) {
  __shared__ float tile[NWAVES][CHUNK][FW_P];         // 36 KB, per-wave slices
  __shared__ float red[NWAVES][NHEADS][10];

  const int tid  = threadIdx.x;
  const int wave = tid >> 5;
  const int lane = tid & 31;
  const int lrow = lane & 15;          // M (or N) index within tile
  const int lkhi = (lane >> 4) << 1;   // K pair base: 0 or 2
  const int rhi  = (lane >> 4) << 3;   // C-row offset: 0 or 8

  // uniform q for this step
  float q[FW];
#pragma unroll
  for (int j = 0; j < FW; ++j) q[j] = ws[WS_Q + j];

  // B fragments of W^T (B[k][n] = W[n][k]) for both 16-col halves, 8 K-steps
  v2f bkf[2][8], bvf[2][8];
#pragma unroll
  for (int nh = 0; nh < 2; ++nh) {
#pragma unroll
    for (int s = 0; s < 8; ++s) {
      bkf[nh][s] = *(const v2f*)&Wk[(16 * nh + lrow) * FW + 4 * s + lkhi];
      bvf[nh][s] = *(const v2f*)&Wv[(16 * nh + lrow) * FW + 4 * s + lkhi];
    }
  }
  float biask[2], biasv[2];
  biask[0] = bk[lrow]; biask[1] = bk[16 + lrow];
  biasv[0] = bv[lrow]; biasv[1] = bv[16 + lrow];

  // online-softmax running state (per lane, per head)
  float m[NHEADS], l[NHEADS], acc[FW];
#pragma unroll
  for (int h = 0; h < NHEADS; ++h) { m[h] = -3.0e38f; l[h] = 0.f; }
#pragma unroll
  for (int j = 0; j < FW; ++j) acc[j] = 0.f;

  const int wgid = blockIdx.x * NWAVES + wave;
  const int base = wgid * POS_PER_WAVE;

  for (int it = 0; it < NITER; ++it) {
    const int cb = base + it * CHUNK;
    if (it + 1 < NITER)   // pull next chunk toward L2/L0 (global_prefetch_b8)
      __builtin_prefetch(&zp[(cb + CHUNK + lane) * FW], 0, 1);

    // ---- K projection: keff tiles -> LDS (exact fp32 via WMMA f32 16x16x4)
#pragma unroll
    for (int t = 0; t < 2; ++t) {
      v2f a[8];
#pragma unroll
      for (int s = 0; s < 8; ++s)
        a[s] = *(const v2f*)&zp[(cb + 16 * t + lrow) * FW + 4 * s + lkhi];
#pragma unroll
      for (int nh = 0; nh < 2; ++nh) {
        v8f c = {};
#pragma unroll
        for (int s = 0; s < 8; ++s)
          c = __builtin_amdgcn_wmma_f32_16x16x4_f32(
              false, a[s], false, bkf[nh][s], (short)0, c, false, false);
#pragma unroll
        for (int r = 0; r < 8; ++r)
          tile[wave][16 * t + r + rhi][16 * nh + lrow] = c[r] + biask[nh];
      }
    }

    // ---- logits + online softmax state (lane p = cb + lane)
    const int p = cb + lane;
    int ridx = p + RELMAX; if (ridx > 2 * RELMAX) ridx = 2 * RELMAX;
    const float* kef = &tile[wave][lane][0];
    float w[NHEADS];
#pragma unroll
    for (int h = 0; h < NHEADS; ++h) {
      float d = 0.f;
#pragma unroll
      for (int u = 0; u < 8; ++u) d += q[8 * h + u] * kef[8 * h + u];
      const float logit = 0.5f * d + ws[WS_RTAB + ridx * NHEADS + h];
      const float nm = fmaxf(m[h], logit);
      const float sc = __expf(m[h] - nm);
      const float wv = __expf(logit - nm);
      l[h] = l[h] * sc + wv;
      m[h] = nm;
      w[h] = wv;
#pragma unroll
      for (int u = 0; u < 8; ++u) acc[8 * h + u] *= sc;
    }

    // ---- V projection (reuse same LDS slice; same-wave DS ops are in-order)
#pragma unroll
    for (int t = 0; t < 2; ++t) {
      v2f a[8];
#pragma unroll
      for (int s = 0; s < 8; ++s)
        a[s] = *(const v2f*)&zp[(cb + 16 * t + lrow) * FW + 4 * s + lkhi];
#pragma unroll
      for (int nh = 0; nh < 2; ++nh) {
        v8f c = {};
#pragma unroll
        for (int s = 0; s < 8; ++s)
          c = __builtin_amdgcn_wmma_f32_16x16x4_f32(
              false, a[s], false, bvf[nh][s], (short)0, c, false, false);
#pragma unroll
        for (int r = 0; r < 8; ++r)
          tile[wave][16 * t + r + rhi][16 * nh + lrow] = c[r] + biasv[nh];
      }
    }
    const float* vr = &tile[wave][lane][0];
#pragma unroll
    for (int h = 0; h < NHEADS; ++h)
#pragma unroll
      for (int u = 0; u < 8; ++u) acc[8 * h + u] += w[h] * vr[8 * h + u];
  }

  // ---- cross-lane online-softmax merge (butterfly over 32 lanes)
#pragma unroll
  for (int off = 16; off > 0; off >>= 1) {
#pragma unroll
    for (int h = 0; h < NHEADS; ++h) {
      const float mo = __shfl_xor(m[h], off, 32);
      const float lo = __shfl_xor(l[h], off, 32);
      const float nm = fmaxf(m[h], mo);
      const float sa = __expf(m[h] - nm);
      const float sb = __expf(mo - nm);
      l[h] = l[h] * sa + lo * sb;
#pragma unroll
      for (int u = 0; u < 8; ++u) {
        const float ao = __shfl_xor(acc[8 * h + u], off, 32);
        acc[8 * h + u] = acc[8 * h + u] * sa + ao * sb;
      }
      m[h] = nm;
    }
  }
  if (lane == 0) {
#pragma unroll
    for (int h = 0; h < NHEADS; ++h) {
      red[wave][h][0] = m[h];
      red[wave][h][1] = l[h];
#pragma unroll
      for (int u = 0; u < 8; ++u) red[wave][h][2 + u] = acc[8 * h + u];
    }
  }
  __syncthreads();
  // ---- merge 8 wave partials -> one block partial per head
  if (tid < NHEADS) {
    const int h = tid;
    float M = red[0][h][0], L = red[0][h][1], A[8];
#pragma unroll
    for (int u = 0; u < 8; ++u) A[u] = red[0][h][2 + u];
    for (int wv2 = 1; wv2 < NWAVES; ++wv2) {
      const float mo = red[wv2][h][0];
      const float nm = fmaxf(M, mo);
      const float sa = __expf(M - nm);
      const float sb = __expf(mo - nm);
      L = L * sa + red[wv2][h][1] * sb;
#pragma unroll
      for (int u = 0; u < 8; ++u) A[u] = A[u] * sa + red[wv2][h][2 + u] * sb;
      M = nm;
    }
    float* pp = &ws[WS_PART + (blockIdx.x * NHEADS + h) * 10];
    pp[0] = M; pp[1] = L;
#pragma unroll
    for (int u = 0; u < 8; ++u) pp[2 + u] = A[u];
  }
}

// ------------------------------------------------------------------
// combine: merge block partials, Wo projection, z update, mish, LN,
//          next-step q and rel table; write d_out on last step.
// ------------------------------------------------------------------
__global__ __launch_bounds__(256) void pil_combine(
    const float* __restrict__ Wo, const float* __restrict__ bo,
    const float* __restrict__ Wq, const float* __restrict__ bq,
    const float* __restrict__ rel, const float* __restrict__ coupling,
    const float* __restrict__ norm_scale,
    float* __restrict__ ws, float* __restrict__ out, int step) {
  __shared__ float so[FW];   // attention output (interleaved re/im)
  __shared__ float sz[FW];   // post-mish z
  __shared__ float sq[FW];   // next q
  __shared__ float smu[2], srs[2];
  const int t = threadIdx.x;

  if (t < NHEADS) {
    const int h = t;
    float M = -3.0e38f, L = 0.f, A[8] = {0, 0, 0, 0, 0, 0, 0, 0};
    for (int b = 0; b < NBLK; ++b) {
      const float* pb = &ws[WS_PART + (b * NHEADS + h) * 10];
      const float mo = pb[0];
      const float nm = fmaxf(M, mo);
      const float sa = __expf(M - nm);
      const float sb = __expf(mo - nm);
      L = L * sa + pb[1] * sb;
#pragma unroll
      for (int u = 0; u < 8; ++u) A[u] = A[u] * sa + pb[2 + u] * sb;
      M = nm;
    }
    const float inv = 1.f / L;
#pragma unroll
    for (int u = 0; u < 8; ++u) so[8 * h + u] = A[u] * inv;
  }
  __syncthreads();
  if (t < FW) {                       // mod = Wo*out + bo ; z += c*mod ; mish
    float mod = bo[t];
#pragma unroll
    for (int i = 0; i < FW; ++i) mod += Wo[t * FW + i] * so[i];
    float zj = ws[WS_Z + t] + coupling[step] * mod;
    sz[t] = mishf(zj);
  }
  __syncthreads();
  if (t < 2) {                        // LN stats over 16 reals / 16 imags
    float mu = 0.f;
#pragma unroll
    for (int k = 0; k < DIMC; ++k) mu += sz[2 * k + t];
    mu *= (1.f / DIMC);
    float var = 0.f;
#pragma unroll
    for (int k = 0; k < DIMC; ++k) {
      const float d = sz[2 * k + t] - mu;
      var += d * d;
    }
    var *= (1.f / DIMC);
    smu[t] = mu;
    srs[t] = rsqrtf(var + 1e-5f);
  }
  __syncthreads();
  if (t < FW) {
    const float zn = (sz[t] - smu[t & 1]) * srs[t & 1] * norm_scale[0];
    ws[WS_Z + t] = zn;
    sz[t] = zn;
    if (step == 7) out[t] = zn;       // complex[16] as interleaved [16][2]
  }
  __syncthreads();
  if (t < FW) {                       // next-step q
    float a = bq[t];
#pragma unroll
    for (int i = 0; i < FW; ++i) a += Wq[t * FW + i] * sz[i];
    sq[t] = a;
    ws[WS_Q + t] = a;
  }
  __syncthreads();
  for (int e = t; e < (2 * RELMAX + 1) * NHEADS; e += 256) {
    const int r = e >> 2, h = e & 3;
    float a = 0.f;
#pragma unroll
    for (int u = 0; u < 8; ++u) a += sq[8 * h + u] * rel[r * FW + 8 * h + u];
    ws[WS_RTAB + e] = 0.5f * a;
  }
}

// ------------------------------------------------------------------
extern "C" void kernel_launch(void* const* d_in, const int* in_sizes, int n_in,
                              void* d_out, int out_size, void* d_ws, size_t ws_size,
                              hipStream_t stream) {
  const float* cand = (const float*)d_in[0];
  const float* zp   = (const float*)d_in[1];
  const float* Wq   = (const float*)d_in[2];
  const float* bq   = (const float*)d_in[3];
  const float* Wk   = (const float*)d_in[4];
  const float* bk   = (const float*)d_in[5];
  const float* Wv   = (const float*)d_in[6];
  const float* bv   = (const float*)d_in[7];
  const float* Wo   = (const float*)d_in[8];
  const float* bo   = (const float*)d_in[9];
  const float* rel  = (const float*)d_in[10];
  const float* coupling   = (const float*)d_in[11];
  const float* norm_scale = (const float*)d_in[12];
  float* ws  = (float*)d_ws;
  float* out = (float*)d_out;

  pil_init<<<1, 256, 0, stream>>>(cand, Wq, bq, rel, ws);
  for (int m = 0; m < 8; ++m) {
    pil_scan<<<NBLK, TPB, 0, stream>>>(zp, Wk, bk, Wv, bv, ws);
    pil_combine<<<1, 256, 0, stream>>>(Wo, bo, Wq, bq, rel, coupling,
                                       norm_scale, ws, out, m);
  }
}